// GatedLinearAttention_38242388803945
// MI455X (gfx1250) — compile-verified
//
#include <hip/hip_runtime.h>

// ---------------- problem constants ----------------
#define BB   2
#define TT   2048
#define DD   1024
#define HH   4
#define DKC  1024
#define DVC  2048
#define HKC  256
#define HVC  512
#define BT   (BB*TT)        // 4096
#define NCH  (TT/64)        // 32 chunks of 64
#define NVB  (HVC/64)       // 8 v-blocks of 64 per head

typedef float v2f __attribute__((ext_vector_type(2)));
typedef float v8f __attribute__((ext_vector_type(8)));

__device__ __forceinline__ v8f wmma4(v8f c, v2f a, v2f b) {
  // D(16x16,f32) = A(16x4,f32) * B(4x16,f32) + C
  return __builtin_amdgcn_wmma_f32_16x16x4_f32(false, a, false, b, (short)0, c, false, false);
}
__device__ __forceinline__ v8f v8z() {
  v8f z = {0.f,0.f,0.f,0.f,0.f,0.f,0.f,0.f};
  return z;
}

// ---------------- generic fp32 WMMA GEMM: C = A[MxK] @ B[KxN] -------------
// block tile 64x256, 8 waves (2x4), each wave 32x64 (2x4 16x16 tiles)
// op: 0 = none, 1 = scale by p0, 2 = silu
__global__ __launch_bounds__(256) void gemm_f32_wmma(
    const float* __restrict__ A, const float* __restrict__ B,
    float* __restrict__ C, int M, int N, int K, int op, float p0) {
  __shared__ float As[64][36];     // stride 36 (≡4 mod 32): conflict-free
  __shared__ float Bs[32][260];    // stride 260 (≡4 mod 64)
  const int tid  = threadIdx.x;
  const int lane = tid & 31, w = tid >> 5;
  const int m  = lane & 15, hi = lane >> 4;
  const int wm = w & 1, wn = w >> 1;        // 2 x 4 wave grid
  const int rowBase = blockIdx.x * 64;
  const int colBase = blockIdx.y * 256;

  v8f acc[2][4];
#pragma unroll
  for (int ti = 0; ti < 2; ++ti)
#pragma unroll
    for (int tj = 0; tj < 4; ++tj) acc[ti][tj] = v8z();

  for (int k0 = 0; k0 < K; k0 += 32) {
#pragma unroll
    for (int i = 0; i < 8; ++i) {           // 64x32 = 2048 = 256*8
      int idx = tid + i * 256;
      int r = idx >> 5, c = idx & 31;
      As[r][c] = A[(rowBase + r) * K + k0 + c];
    }
#pragma unroll
    for (int i = 0; i < 32; ++i) {          // 32x256 = 8192 = 256*32
      int idx = tid + i * 256;
      int r = idx >> 8, c = idx & 255;
      Bs[r][c] = B[(k0 + r) * N + colBase + c];
    }
    __syncthreads();
    // prefetch next K-panel into cache hierarchy (global_prefetch_b8)
    if (k0 + 32 < K) {
      __builtin_prefetch(A + (rowBase + (tid & 63)) * K + k0 + 32, 0, 0);
      __builtin_prefetch(B + (size_t)(k0 + 32 + (tid & 31)) * N + colBase + ((tid >> 5) << 5), 0, 0);
    }
#pragma unroll
    for (int kb = 0; kb < 32; kb += 4) {
      v2f a[2], b[4];
#pragma unroll
      for (int ti = 0; ti < 2; ++ti) {
        int ar = wm * 32 + ti * 16 + m;
        a[ti].x = As[ar][kb + 2 * hi];
        a[ti].y = As[ar][kb + 2 * hi + 1];
      }
#pragma unroll
      for (int tj = 0; tj < 4; ++tj) {
        int bc = wn * 64 + tj * 16 + m;
        b[tj].x = Bs[kb + 2 * hi][bc];
        b[tj].y = Bs[kb + 2 * hi + 1][bc];
      }
#pragma unroll
      for (int ti = 0; ti < 2; ++ti)
#pragma unroll
        for (int tj = 0; tj < 4; ++tj)
          acc[ti][tj] = wmma4(acc[ti][tj], a[ti], b[tj]);
    }
    __syncthreads();
  }
#pragma unroll
  for (int ti = 0; ti < 2; ++ti)
#pragma unroll
    for (int tj = 0; tj < 4; ++tj)
#pragma unroll
      for (int r = 0; r < 8; ++r) {
        int gr = rowBase + wm * 32 + ti * 16 + r + 8 * hi;
        int gc = colBase + wn * 64 + tj * 16 + m;
        float v = acc[ti][tj][r];
        if (op == 1) v *= p0;
        else if (op == 2) v = v / (1.f + __expf(-v));   // silu
        C[gr * N + gc] = v;
      }
}

// ---------------- low-rank gate part 1: X1 = x @ gk_w1  [BT x 16] ---------
__global__ __launch_bounds__(64) void lowrank1_kernel(
    const float* __restrict__ x, const float* __restrict__ W1, float* __restrict__ X1) {
  __shared__ float red[64];
  const int row = blockIdx.x;
  const int col = threadIdx.x & 15;
  const int seg = threadIdx.x >> 4;   // 0..3, 256 elems each
  float s = 0.f;
  const float* xr = x + row * DD + seg * 256;
#pragma unroll 4
  for (int kk = 0; kk < 256; ++kk) s += xr[kk] * W1[(seg * 256 + kk) * 16 + col];
  red[threadIdx.x] = s;
  __syncthreads();
  if (threadIdx.x < 16)
    X1[row * 16 + threadIdx.x] =
        red[threadIdx.x] + red[threadIdx.x + 16] + red[threadIdx.x + 32] + red[threadIdx.x + 48];
}

// ---------------- gate: G = log_sigmoid(X1 @ gk_w2 + b) / 16 --------------
__global__ __launch_bounds__(256) void gate_kernel(
    const float* __restrict__ X1, const float* __restrict__ W2,
    const float* __restrict__ bias, float* __restrict__ G) {
  __shared__ float xr[16];
  const int row = blockIdx.x;
  const int col = blockIdx.y * 256 + threadIdx.x;
  if (threadIdx.x < 16) xr[threadIdx.x] = X1[row * 16 + threadIdx.x];
  __syncthreads();
  float s = bias[col];
#pragma unroll
  for (int r = 0; r < 16; ++r) s += xr[r] * W2[r * DKC + col];
  float ls = fminf(s, 0.f) - log1pf(__expf(-fabsf(s)));   // stable log-sigmoid
  G[row * DKC + col] = ls * (1.f / 16.f);
}

// ---------------- chunked GLA scan (fp32 WMMA) ----------------------------
// grid: B*H*NVB = 64 blocks; 256 threads (8 waves); dynamic LDS ~233 KB
__global__ __launch_bounds__(256) void gla_chunk_kernel(
    const float* __restrict__ Q, const float* __restrict__ K,
    const float* __restrict__ V, const float* __restrict__ G,
    float* __restrict__ O) {
  extern __shared__ float sm[];
  float* S  = sm;                  // [256][68]  state, persistent across chunks
  float* Qt = S  + 256 * 68;       // [64][260]  q * exp(B)
  float* Kt = Qt + 64 * 260;       // [64][260]  k * exp(-B)
  float* Vt = Kt + 64 * 260;       // [64][68]
  float* At = Vt + 64 * 68;        // [64][68]   tril attention
  float* eg = At + 64 * 68;        // [256]      exp(B_C) per k-dim

  const int tid  = threadIdx.x;
  const int lane = tid & 31, w = tid >> 5;
  const int m = lane & 15, hi = lane >> 4;
  const int vb = blockIdx.x & 7;
  const int h  = (blockIdx.x >> 3) & 3;
  const int b  = blockIdx.x >> 5;
  const int rowOff = b * TT;

  for (int i = tid; i < 256 * 68; i += 256) S[i] = 0.f;
  __syncthreads();

  for (int c = 0; c < NCH; ++c) {
    const int t0 = c * 64;
    // ---- phase 1: per-column cumulative decays; build Qt, Kt, eg ----
    {
      const int d = tid;
      const float* gq = G + (rowOff + t0) * DKC + h * HKC + d;
      const float* qq = Q + (rowOff + t0) * DKC + h * HKC + d;
      const float* kq = K + (rowOff + t0) * DKC + h * HKC + d;
      float run = 0.f;
      for (int t = 0; t < 64; ++t) {
        run += gq[t * DKC];
        Qt[t * 260 + d] = qq[t * DKC] * __expf(run);
        Kt[t * 260 + d] = kq[t * DKC] * __expf(-run);
      }
      eg[d] = __expf(run);
    }
    // ---- V tile load (64 x 64 slice of this v-block) ----
    for (int idx = tid; idx < 64 * 64; idx += 256) {
      int t = idx >> 6, j = idx & 63;
      Vt[t * 68 + j] = V[(rowOff + t0 + t) * DVC + h * HVC + vb * 64 + j];
    }
    __syncthreads();

    // ---- phase 2: A = tril(Qt @ Kt^T)  [64x64], 16 tiles / 8 waves ----
#pragma unroll
    for (int i = 0; i < 2; ++i) {
      const int tile = w * 2 + i, ti = tile >> 2, tj = tile & 3;
      v8f acc = v8z();
      if (ti >= tj) {                 // wave-uniform; upper tiles all-masked
        for (int kb = 0; kb < HKC; kb += 4) {
          v2f a, bb;
          a.x  = Qt[(ti * 16 + m) * 260 + kb + 2 * hi];
          a.y  = Qt[(ti * 16 + m) * 260 + kb + 2 * hi + 1];
          bb.x = Kt[(tj * 16 + m) * 260 + kb + 2 * hi];      // B = Kt^T
          bb.y = Kt[(tj * 16 + m) * 260 + kb + 2 * hi + 1];
          acc = wmma4(acc, a, bb);
        }
      }
#pragma unroll
      for (int r = 0; r < 8; ++r) {
        int rt = ti * 16 + r + 8 * hi, s = tj * 16 + m;
        At[rt * 68 + s] = (rt >= s) ? acc[r] : 0.f;
      }
    }
    __syncthreads();

    // ---- phase 3: O = A @ V + Qt @ S  [64x64] ----
    // wave owns one ti row-block and two adjacent tj blocks: A-frag reused 2x
    {
      const int ti  = w >> 1;            // 0..3
      const int tjb = (w & 1) * 2;       // 0 or 2
      v8f acc0 = v8z(), acc1 = v8z();
      for (int kb = 0; kb < 64; kb += 4) {            // intra-chunk: A @ V
        v2f a, b0, b1;
        a.x  = At[(ti * 16 + m) * 68 + kb + 2 * hi];
        a.y  = At[(ti * 16 + m) * 68 + kb + 2 * hi + 1];
        b0.x = Vt[(kb + 2 * hi) * 68 + tjb * 16 + m];
        b0.y = Vt[(kb + 2 * hi + 1) * 68 + tjb * 16 + m];
        b1.x = Vt[(kb + 2 * hi) * 68 + (tjb + 1) * 16 + m];
        b1.y = Vt[(kb + 2 * hi + 1) * 68 + (tjb + 1) * 16 + m];
        acc0 = wmma4(acc0, a, b0);
        acc1 = wmma4(acc1, a, b1);
      }
      for (int kb = 0; kb < HKC; kb += 4) {           // inter-chunk: Qt @ S_old
        v2f a, b0, b1;
        a.x  = Qt[(ti * 16 + m) * 260 + kb + 2 * hi];
        a.y  = Qt[(ti * 16 + m) * 260 + kb + 2 * hi + 1];
        b0.x = S[(kb + 2 * hi) * 68 + tjb * 16 + m];
        b0.y = S[(kb + 2 * hi + 1) * 68 + tjb * 16 + m];
        b1.x = S[(kb + 2 * hi) * 68 + (tjb + 1) * 16 + m];
        b1.y = S[(kb + 2 * hi + 1) * 68 + (tjb + 1) * 16 + m];
        acc0 = wmma4(acc0, a, b0);
        acc1 = wmma4(acc1, a, b1);
      }
#pragma unroll
      for (int r = 0; r < 8; ++r) {
        int grow = rowOff + t0 + ti * 16 + r + 8 * hi;
        int gcol = h * HVC + vb * 64 + tjb * 16 + m;
        O[grow * DVC + gcol]      = acc0[r];
        O[grow * DVC + gcol + 16] = acc1[r];
      }
    }
    __syncthreads();

    // ---- phase 4: S = diag(eg) * (S + Kt^T @ V) ----
    // wave owns si ∈ {2w, 2w+1} x all 4 sj: 2 A-frags + 4 B-frags -> 8 wmma
    {
      const int si0 = w * 2, si1 = w * 2 + 1;   // 0..15
      v8f sacc[2][4];
#pragma unroll
      for (int s2 = 0; s2 < 2; ++s2)
#pragma unroll
        for (int sj = 0; sj < 4; ++sj)
#pragma unroll
          for (int r = 0; r < 8; ++r)
            sacc[s2][sj][r] = S[((si0 + s2) * 16 + r + 8 * hi) * 68 + sj * 16 + m];
      for (int kb = 0; kb < 64; kb += 4) {
        v2f a0, a1, bfr[4];
        a0.x = Kt[(kb + 2 * hi) * 260 + si0 * 16 + m];        // A = Kt^T
        a0.y = Kt[(kb + 2 * hi + 1) * 260 + si0 * 16 + m];
        a1.x = Kt[(kb + 2 * hi) * 260 + si1 * 16 + m];
        a1.y = Kt[(kb + 2 * hi + 1) * 260 + si1 * 16 + m];
#pragma unroll
        for (int sj = 0; sj < 4; ++sj) {
          bfr[sj].x = Vt[(kb + 2 * hi) * 68 + sj * 16 + m];
          bfr[sj].y = Vt[(kb + 2 * hi + 1) * 68 + sj * 16 + m];
        }
#pragma unroll
        for (int sj = 0; sj < 4; ++sj) {
          sacc[0][sj] = wmma4(sacc[0][sj], a0, bfr[sj]);
          sacc[1][sj] = wmma4(sacc[1][sj], a1, bfr[sj]);
        }
      }
#pragma unroll
      for (int s2 = 0; s2 < 2; ++s2)
#pragma unroll
        for (int sj = 0; sj < 4; ++sj)
#pragma unroll
          for (int r = 0; r < 8; ++r) {
            int rw = (si0 + s2) * 16 + r + 8 * hi;
            S[rw * 68 + sj * 16 + m] = sacc[s2][sj][r] * eg[rw];
          }
    }
    __syncthreads();
  }
}

// ---------------- group RMSNorm + swish gate (in-place into Gs) -----------
__global__ __launch_bounds__(128) void rmsgate_kernel(
    const float* __restrict__ O, const float* __restrict__ rms_w,
    float* __restrict__ Gs) {
  __shared__ float red[128];
  const int bh  = blockIdx.x;
  const int row = bh >> 2;          // BT index
  const int h   = bh & 3;
  const float* o = O + row * DVC + h * HVC;
  float ss = 0.f;
  for (int j = threadIdx.x; j < HVC; j += 128) { float v = o[j]; ss += v * v; }
  red[threadIdx.x] = ss;
  __syncthreads();
  for (int s = 64; s > 0; s >>= 1) {
    if (threadIdx.x < s) red[threadIdx.x] += red[threadIdx.x + s];
    __syncthreads();
  }
  const float inv = rsqrtf(red[0] * (1.f / HVC) + 1e-5f);
  float* gp = Gs + row * DVC + h * HVC;
  for (int j = threadIdx.x; j < HVC; j += 128)
    gp[j] = o[j] * inv * rms_w[j] * gp[j];   // gp held silu(x@Wg)
}

// ---------------- host driver --------------------------------------------
extern "C" void kernel_launch(void* const* d_in, const int* in_sizes, int n_in,
                              void* d_out, int out_size, void* d_ws, size_t ws_size,
                              hipStream_t stream) {
  const float* x  = (const float*)d_in[0];
  const float* Wq = (const float*)d_in[1];
  const float* Wk = (const float*)d_in[2];
  const float* Wv = (const float*)d_in[3];
  const float* Wg = (const float*)d_in[4];
  const float* W1 = (const float*)d_in[5];
  const float* W2 = (const float*)d_in[6];
  const float* gb = (const float*)d_in[7];
  const float* Wo = (const float*)d_in[8];
  const float* rw = (const float*)d_in[9];
  float* out = (float*)d_out;

  float* ws  = (float*)d_ws;
  float* Qb  = ws;               ws += (size_t)BT * DKC;
  float* Kb  = ws;               ws += (size_t)BT * DKC;
  float* Vb  = ws;               ws += (size_t)BT * DVC;
  float* Gsb = ws;               ws += (size_t)BT * DVC;   // silu(x@Wg), then gated out
  float* Gb  = ws;               ws += (size_t)BT * DKC;   // log-decays
  float* X1  = ws;               ws += (size_t)BT * 16;
  float* Ob  = ws;               ws += (size_t)BT * DVC;   // scan output

  const dim3 blk(256);
  // projections (fp32 WMMA GEMMs), block tile 64x256
  gemm_f32_wmma<<<dim3(BT / 64, DKC / 256), blk, 0, stream>>>(x, Wq, Qb, BT, DKC, DD, 1, 0.0625f); // q * HK^-0.5
  gemm_f32_wmma<<<dim3(BT / 64, DKC / 256), blk, 0, stream>>>(x, Wk, Kb, BT, DKC, DD, 0, 0.f);
  gemm_f32_wmma<<<dim3(BT / 64, DVC / 256), blk, 0, stream>>>(x, Wv, Vb, BT, DVC, DD, 0, 0.f);
  gemm_f32_wmma<<<dim3(BT / 64, DVC / 256), blk, 0, stream>>>(x, Wg, Gsb, BT, DVC, DD, 2, 0.f);    // silu
  // low-rank forget gate
  lowrank1_kernel<<<BT, 64, 0, stream>>>(x, W1, X1);
  gate_kernel<<<dim3(BT, DKC / 256), blk, 0, stream>>>(X1, W2, gb, Gb);
  // chunked GLA scan: 64 blocks, ~233 KB dynamic LDS each (320 KB WGP LDS)
  const size_t glaLds = (size_t)(256 * 68 + 2 * 64 * 260 + 2 * 64 * 68 + 256) * sizeof(float);
  gla_chunk_kernel<<<BB * HH * NVB, blk, glaLds, stream>>>(Qb, Kb, Vb, Gb, Ob);
  // group RMSNorm + output gate
  rmsgate_kernel<<<BT * HH, 128, 0, stream>>>(Ob, rw, Gsb);
  // output projection
  gemm_f32_wmma<<<dim3(BT / 64, DD / 256), blk, 0, stream>>>(Gsb, Wo, out, BT, DD, DVC, 0, 0.f);
}